// GraphNormalization_249108103838
// MI455X (gfx1250) — compile-verified
//
#include <hip/hip_runtime.h>
#include <math.h>

// GraphNorm for N=1e6 nodes, F=128 features, B=2048 graphs.
// Strategy: counting-sort nodes by graph, per-graph one-pass (sum, sumsq)
// reduction using V_WMMA_F32_16X16X4_F32 with an all-ones A matrix (exact
// f32 column sums), fold everything into per-(graph,feature) scale/bias,
// then a float4 streaming normalize pass. HBM traffic ~1.5 GB (~65us roof).
//
// Round-3 change: straight-line (unrolled) LDS staging of node indices so
// all 4 index loads per thread are issued before any wait (round 2 staged
// one index per serialized load->wait->ds_store round trip).

typedef __attribute__((ext_vector_type(2))) float v2f;
typedef __attribute__((ext_vector_type(8))) float v8f;

#define FEAT 128
#define FC4  32     // float4 chunks per feature row
#define EPSV 1e-3f
#define CH   1024   // node indices staged in LDS per chunk (4KB)

__global__ __launch_bounds__(256) void gn_zero_kernel(int* __restrict__ counts, int B) {
    int i = blockIdx.x * 256 + threadIdx.x;
    if (i < B) counts[i] = 0;
}

__global__ __launch_bounds__(256) void gn_hist_kernel(const int* __restrict__ gid,
                                                      int* __restrict__ counts, int N) {
    int n = blockIdx.x * 256 + threadIdx.x;
    if (n < N) atomicAdd(&counts[gid[n]], 1);
}

// Single-block exclusive scan over B (<= 4096) counts -> offsets[0..B], cursor copy.
__global__ __launch_bounds__(256) void gn_scan_kernel(const int* __restrict__ counts,
                                                      int* __restrict__ offsets,
                                                      int* __restrict__ cursor, int B) {
    __shared__ int part[256];
    int t = threadIdx.x;
    int per = (B + 255) / 256;            // 8 for B=2048
    int base = t * per;
    int local[16];
    int s = 0;
    for (int k = 0; k < per && k < 16; ++k) {
        int v = (base + k < B) ? counts[base + k] : 0;
        local[k] = v;
        s += v;
    }
    part[t] = s;
    __syncthreads();
    if (t == 0) {
        int run = 0;
        for (int i = 0; i < 256; ++i) { int v = part[i]; part[i] = run; run += v; }
        offsets[B] = run;
    }
    __syncthreads();
    int run = part[t];
    for (int k = 0; k < per && k < 16; ++k) {
        if (base + k < B) {
            offsets[base + k] = run;
            cursor[base + k]  = run;
            run += local[k];
        }
    }
}

__global__ __launch_bounds__(256) void gn_scatter_kernel(const int* __restrict__ gid,
                                                         int* __restrict__ cursor,
                                                         int* __restrict__ node_order, int N) {
    int n = blockIdx.x * 256 + threadIdx.x;
    if (n < N) {
        int g = gid[n];
        int p = atomicAdd(&cursor[g], 1);
        node_order[p] = n;
    }
}

// One block (256 thr = 8 waves) per graph. Wave w owns features [16w, 16w+16).
// Per WMMA the wave gathers 4 node rows x 16 features into the B operand of
// V_WMMA_F32_16X16X4_F32 with A = ones(16x4): every row of D becomes the
// exact f32 column sums. Second accumulator takes B*B for sum-of-squares.
// Node element-offsets (node*FEAT) are staged in LDS to keep the gather loop
// branch-free with many loads outstanding.
__global__ __launch_bounds__(256) void gn_stats_wmma_kernel(
        const float* __restrict__ values,
        const int* __restrict__ offsets,
        const int* __restrict__ node_order,
        const float* __restrict__ gamma,
        const float* __restrict__ beta,
        const float* __restrict__ alpha,
        float* __restrict__ scale,
        float* __restrict__ bias) {
    __shared__ int soff[CH];

    int b    = blockIdx.x;
    int beg  = offsets[b];
    int end  = offsets[b + 1];
    int tid  = threadIdx.x;
    int lane = tid & 31;
    int wave = tid >> 5;
    int fl   = wave * 16 + (lane & 15);   // feature this lane owns (column N = lane&15)
    int half2 = (lane >> 4) << 1;         // lanes 0-15 take nodes {0,1}, 16-31 take {2,3}

    v8f sacc = {};
    v8f qacc = {};
    v2f ones = {1.0f, 1.0f};

    for (int cs = beg; cs < end; cs += CH) {
        int m = end - cs;
        if (m > CH) m = CH;

        // Straight-line cooperative stage: 4 index loads issued back-to-back
        // (addresses clamped in-bounds; cs < end <= N), conditional stores.
        {
            int  i0 = tid, i1 = tid + 256, i2 = tid + 512, i3 = tid + 768;
            bool k0 = i0 < m, k1 = i1 < m, k2 = i2 < m, k3 = i3 < m;
            int  n0 = node_order[cs + (k0 ? i0 : 0)];
            int  n1 = node_order[cs + (k1 ? i1 : 0)];
            int  n2 = node_order[cs + (k2 ? i2 : 0)];
            int  n3 = node_order[cs + (k3 ? i3 : 0)];
            if (k0) soff[i0] = n0 * FEAT;
            if (k1) soff[i1] = n1 * FEAT;
            if (k2) soff[i2] = n2 * FEAT;
            if (k3) soff[i3] = n3 * FEAT;
        }
        __syncthreads();

        int nfull = m & ~3;
        #pragma unroll 4
        for (int i = 0; i < nfull; i += 4) {
            int ia = i + half2;
            int o0 = soff[ia];
            int o1 = soff[ia + 1];
            float v0 = values[o0 + fl];
            float v1 = values[o1 + fl];
            v2f bv = {v0, v1};
            v2f bq = {v0 * v0, v1 * v1};
            sacc = __builtin_amdgcn_wmma_f32_16x16x4_f32(false, ones, false, bv,
                                                         (short)0, sacc, false, false);
            qacc = __builtin_amdgcn_wmma_f32_16x16x4_f32(false, ones, false, bq,
                                                         (short)0, qacc, false, false);
        }

        // Tail (<= 3 nodes): zero-pad inactive slots (zeros add nothing).
        if (nfull < m) {
            int ia = nfull + half2;
            float v0 = (ia     < m) ? values[soff[ia]     + fl] : 0.0f;
            float v1 = (ia + 1 < m) ? values[soff[ia + 1] + fl] : 0.0f;
            v2f bv = {v0, v1};
            v2f bq = {v0 * v0, v1 * v1};
            sacc = __builtin_amdgcn_wmma_f32_16x16x4_f32(false, ones, false, bv,
                                                         (short)0, sacc, false, false);
            qacc = __builtin_amdgcn_wmma_f32_16x16x4_f32(false, ones, false, bq,
                                                         (short)0, qacc, false, false);
        }
        __syncthreads();
    }

    // D row M=0 lives in acc[0]: lane j (j<16) holds column N=j; lanes >=16
    // hold row M=8 which equals the same column sums (A all-ones).
    float s = sacc[0];
    float q = qacc[0];
    if (lane < 16) {
        int cnt = end - beg;
        if (cnt < 1) cnt = 1;
        float rc  = 1.0f / (float)cnt;
        float m   = s * rc;              // per-graph mean
        float ms  = alpha[fl] * m;       // mean-shifted
        float var = q * rc - 2.0f * ms * m + ms * ms;   // E[(x-ms)^2]
        float inv = 1.0f / sqrtf(var + EPSV);
        float sc  = gamma[fl] * inv;
        scale[b * FEAT + fl] = sc;
        bias[b * FEAT + fl]  = beta[fl] - ms * sc;
    }
}

__global__ __launch_bounds__(256) void gn_normalize_kernel(
        const float* __restrict__ values,
        const int* __restrict__ gid,
        const float* __restrict__ scale,
        const float* __restrict__ bias,
        float* __restrict__ out, int N) {
    int idx = blockIdx.x * 256 + threadIdx.x;
    int n = idx >> 5;          // node
    int c = idx & 31;          // float4 chunk within the 128-feature row
    if (n >= N) return;
    int g = gid[n];
    const float4* v4 = (const float4*)values;
    const float4* s4 = (const float4*)scale;
    const float4* b4 = (const float4*)bias;
    float4*       o4 = (float4*)out;
    float4 v  = v4[(size_t)n * FC4 + c];
    float4 s  = s4[(size_t)g * FC4 + c];
    float4 bb = b4[(size_t)g * FC4 + c];
    float4 o;
    o.x = v.x * s.x + bb.x;
    o.y = v.y * s.y + bb.y;
    o.z = v.z * s.z + bb.z;
    o.w = v.w * s.w + bb.w;
    o4[(size_t)n * FC4 + c] = o;
}

extern "C" void kernel_launch(void* const* d_in, const int* in_sizes, int n_in,
                              void* d_out, int out_size, void* d_ws, size_t ws_size,
                              hipStream_t stream) {
    const float* values = (const float*)d_in[0];
    const int*   gid    = (const int*)d_in[1];
    // d_in[2] = reference_ids (arange) -> only its length B is needed
    const float* gamma  = (const float*)d_in[3];
    const float* beta   = (const float*)d_in[4];
    const float* alpha  = (const float*)d_in[5];
    float*       out    = (float*)d_out;

    int N = in_sizes[1];
    int B = in_sizes[2];

    // Workspace carve-out (int/float elements, each region 16B-aligned).
    int* ws_i = (int*)d_ws;
    size_t pc = 0;
    int* counts     = ws_i + pc; pc += ((size_t)B + 3) & ~(size_t)3;
    int* offsets    = ws_i + pc; pc += ((size_t)B + 1 + 3) & ~(size_t)3;
    int* cursor     = ws_i + pc; pc += ((size_t)B + 3) & ~(size_t)3;
    int* node_order = ws_i + pc; pc += ((size_t)N + 3) & ~(size_t)3;
    float* scale    = (float*)(ws_i + pc); pc += ((size_t)B * FEAT + 3) & ~(size_t)3;
    float* bias     = (float*)(ws_i + pc);

    int nbB = (B + 255) / 256;
    int nbN = (N + 255) / 256;

    gn_zero_kernel<<<nbB, 256, 0, stream>>>(counts, B);
    gn_hist_kernel<<<nbN, 256, 0, stream>>>(gid, counts, N);
    gn_scan_kernel<<<1, 256, 0, stream>>>(counts, offsets, cursor, B);
    gn_scatter_kernel<<<nbN, 256, 0, stream>>>(gid, cursor, node_order, N);
    gn_stats_wmma_kernel<<<B, 256, 0, stream>>>(values, offsets, node_order,
                                                gamma, beta, alpha, scale, bias);
    long long total = (long long)N * FC4;
    int nbE = (int)((total + 255) / 256);
    gn_normalize_kernel<<<nbE, 256, 0, stream>>>(values, gid, scale, bias, out, N);
}